// StandardTripletAnalogyLoss_9096740733656
// MI455X (gfx1250) — compile-verified
//
#include <hip/hip_runtime.h>
#include <math.h>

#define EPS    1e-6f
#define MARGIN 1.0f

typedef float v2f __attribute__((ext_vector_type(2)));
typedef float v8f __attribute__((ext_vector_type(8)));

// ---------------------------------------------------------------------------
// Kernel 1: one wave32 per row. Streams e/a/b/c with float4 (global_load_b128)
// loads, accumulates squared diffs per lane in f32 (exact fmaf), cross-lane
// reduces with shfl_xor, writes per-row hinge loss to workspace.
// Memory-bound: 128 MiB @ 23.3 TB/s ~ 5.8 us floor.
// ---------------------------------------------------------------------------
__global__ __launch_bounds__(256) void triplet_rows_kernel(
    const float* __restrict__ ex, const float* __restrict__ ca,
    const float* __restrict__ cb, const float* __restrict__ cc,
    const int*   __restrict__ idx, float* __restrict__ row_loss, int D)
{
    const int wave = (int)((blockIdx.x * blockDim.x + threadIdx.x) >> 5); // row id
    const int lane = (int)(threadIdx.x & 31);
    const long long rb = (long long)wave * (long long)D;

    const float4* e4 = (const float4*)(ex + rb);
    const float4* a4 = (const float4*)(ca + rb);
    const float4* b4 = (const float4*)(cb + rb);
    const float4* c4 = (const float4*)(cc + rb);

    float sa = 0.0f, sb = 0.0f, sc = 0.0f;
    const int nvec = D >> 2; // 512 float4 per row

    for (int i = lane; i < nvec; i += 32) {
        float4 e = e4[i];
        float4 a = a4[i];
        float4 b = b4[i];
        float4 c = c4[i];
        float d;
        d = e.x - a.x + EPS; sa = fmaf(d, d, sa);
        d = e.y - a.y + EPS; sa = fmaf(d, d, sa);
        d = e.z - a.z + EPS; sa = fmaf(d, d, sa);
        d = e.w - a.w + EPS; sa = fmaf(d, d, sa);
        d = e.x - b.x + EPS; sb = fmaf(d, d, sb);
        d = e.y - b.y + EPS; sb = fmaf(d, d, sb);
        d = e.z - b.z + EPS; sb = fmaf(d, d, sb);
        d = e.w - b.w + EPS; sb = fmaf(d, d, sb);
        d = e.x - c.x + EPS; sc = fmaf(d, d, sc);
        d = e.y - c.y + EPS; sc = fmaf(d, d, sc);
        d = e.z - c.z + EPS; sc = fmaf(d, d, sc);
        d = e.w - c.w + EPS; sc = fmaf(d, d, sc);
    }

    // wave32 cross-lane tree reduction (deterministic order)
    for (int off = 16; off > 0; off >>= 1) {
        sa += __shfl_xor(sa, off, 32);
        sb += __shfl_xor(sb, off, 32);
        sc += __shfl_xor(sc, off, 32);
    }

    if (lane == 0) {
        float da = sqrtf(sa), db = sqrtf(sb), dc = sqrtf(sc);
        int k = idx[wave];
        // d_pos equals the d_neg of the correct choice (same +eps formula),
        // so only 3 distances are needed.
        float dpos = (k == 0) ? da : ((k == 1) ? db : dc);
        float loss = 0.0f;
        if (k != 0) loss += fmaxf(dpos - da + MARGIN, 0.0f);
        if (k != 1) loss += fmaxf(dpos - db + MARGIN, 0.0f);
        if (k != 2) loss += fmaxf(dpos - dc + MARGIN, 0.0f);
        row_loss[wave] = loss;
    }
}

// ---------------------------------------------------------------------------
// Kernel 2: single-wave exact f32 reduction of n (multiple of 64) values using
// chained V_WMMA_F32_16X16X4_F32 as the accumulator:
//   A = 64 loaded values (v2f/lane), B = ones  =>  D[i][j] = rowsum_i(A) + C
// After the loop, summing one column of D (a lane's 8 C VGPRs) gives rows 0-7
// (lanes 0-15) / rows 8-15 (lanes 16-31); one shfl_xor(16) completes the sum.
// EXEC is all-ones throughout the WMMA loop (32 threads, uniform control).
// ---------------------------------------------------------------------------
__global__ __launch_bounds__(32) void wmma_reduce_kernel(
    const float* __restrict__ in, float* __restrict__ out, int n, float scale)
{
    const int lane = (int)(threadIdx.x & 31);
    v8f acc = {};
    v2f ones;
    ones[0] = 1.0f;
    ones[1] = 1.0f;

    for (int base = 0; base < n; base += 64) {
        v2f a = *(const v2f*)(in + base + lane * 2);
        // 8 args: (neg_a, A, neg_b, B, c_mod, C, reuse_a, reuse_b)
        acc = __builtin_amdgcn_wmma_f32_16x16x4_f32(
            /*neg_a=*/false, a, /*neg_b=*/false, ones,
            /*c_mod=*/(short)0, acc, /*reuse_a=*/false, /*reuse_b=*/false);
    }

    float s = acc[0] + acc[1] + acc[2] + acc[3] + acc[4] + acc[5] + acc[6] + acc[7];
    s += __shfl_xor(s, 16, 32);
    if (lane == 0) out[0] = s * scale;
}

extern "C" void kernel_launch(void* const* d_in, const int* in_sizes, int n_in,
                              void* d_out, int out_size, void* d_ws, size_t ws_size,
                              hipStream_t stream)
{
    const float* ex  = (const float*)d_in[0];
    const float* ca  = (const float*)d_in[1];
    const float* cb  = (const float*)d_in[2];
    const float* cc  = (const float*)d_in[3];
    const int*   idx = (const int*)d_in[4];

    const int B = in_sizes[4];            // 4096 rows
    const int D = in_sizes[0] / B;        // 2048 cols

    float* row_loss = (float*)d_ws;       // B floats of scratch

    // one wave (32 threads) per row -> B*32 threads total
    const int threads = 256;
    const int blocks  = (B * 32 + threads - 1) / threads;
    triplet_rows_kernel<<<blocks, threads, 0, stream>>>(ex, ca, cb, cc, idx, row_loss, D);

    // B = 4096 is a multiple of 64 -> exact single-wave WMMA reduction
    wmma_reduce_kernel<<<1, 32, 0, stream>>>(row_loss, (float*)d_out, B,
                                             1.0f / (float)(B * 2));
}